// Raymarcher_49838800503335
// MI455X (gfx1250) — compile-verified
//
#include <hip/hip_runtime.h>
#include <cstdint>

// ---------------------------------------------------------------------------
// Raymarcher for MI455X (gfx1250).
// Bandwidth-bound: ~335MB streaming stores + L2-resident 8MB grid gather.
//  - TDM (tensor_load_to_lds) stages per-block ray data into LDS async.
//  - All output stores are non-temporal (th:TH_STORE_NT) so the 335MB write
//    stream doesn't evict the 8MB density grid from the 192MB L2.
//  - Perfectly coalesced global_store_b128 streams (48B/lane xyz, 16B z/valid).
// ---------------------------------------------------------------------------

#define N_RAYS 131072
#define M_SAMP 128
#define GRID_R 128
#define RAYS_PER_BLOCK 8          // 256 threads = 8 waves, 1 wave per ray
#define THREADS 256

typedef unsigned int v4u __attribute__((ext_vector_type(4)));
typedef int          v8i __attribute__((ext_vector_type(8)));
typedef int          v4i __attribute__((ext_vector_type(4)));
typedef float        v4f __attribute__((ext_vector_type(4)));

__global__ __launch_bounds__(THREADS) void raymarch_kernel(
    const float* __restrict__ rays,      // [N_RAYS, 6]  (d.xyz, o.xyz)
    const float* __restrict__ dgrid,     // [GRID^3]
    float* __restrict__ out_xyz,         // [N_RAYS, M, 3]
    float* __restrict__ out_z,           // [N_RAYS, M]
    float* __restrict__ out_valid)       // [N_RAYS, M]  (0.0 / 1.0)
{
    __shared__ float s_rays[RAYS_PER_BLOCK * 6];   // 192 bytes

    const int block0 = blockIdx.x * RAYS_PER_BLOCK;

    // ---- TDM: async DMA 48 dwords of ray data global -> LDS ----------------
    if (threadIdx.x == 0) {
        const uint64_t ga  = (uint64_t)(uintptr_t)(rays + (size_t)block0 * 6);
        const uint32_t lds = (uint32_t)(uintptr_t)(&s_rays[0]); // low 32b of flat = LDS offset
        v4u g0;
        g0[0] = 1u;                                        // count=1, user mode
        g0[1] = lds;                                       // lds_addr (bytes)
        g0[2] = (uint32_t)(ga & 0xFFFFFFFFu);              // global_addr[31:0]
        g0[3] = (uint32_t)((ga >> 32) & 0x01FFFFFFu)       // global_addr[56:32]
              | (2u << 30);                                // type = 2 ("image")
        v8i g1;
        g1[0] = (2 << 16);          // wg_mask=0 (not in cluster), data_size=4B
        g1[1] = (48 << 16);         // tensor_dim0[15:0] = 48 dwords
        g1[2] = (1  << 16);         // tensor_dim0 hi = 0, tensor_dim1 = 1
        g1[3] = (48 << 16);         // tensor_dim1 hi = 0, tile_dim0 = 48 (192B, mult of 4B)
        g1[4] = 0;                  // tile_dim1 = 0 (unused -> 1-D), tile_dim2 = 0
        g1[5] = 48;                 // tensor_dim0_stride[31:0]
        g1[6] = 0;                  // stride hi / tensor_dim1_stride lo
        g1[7] = 0;
        v4i z4 = {0, 0, 0, 0};
#if defined(__clang_major__) && (__clang_major__ >= 23)
        v8i z8 = {0, 0, 0, 0, 0, 0, 0, 0};
        __builtin_amdgcn_tensor_load_to_lds(g0, g1, z4, z4, z8, 0);
#else
        __builtin_amdgcn_tensor_load_to_lds(g0, g1, z4, z4, 0);
#endif
        __builtin_amdgcn_s_wait_tensorcnt(0);
    }
    __syncthreads();

    const int lane   = threadIdx.x & 31;   // sample group within ray
    const int rayIdx = threadIdx.x >> 5;   // wave id == ray within block
    const int ray    = block0 + rayIdx;

    // Broadcast ray data from LDS (one bank transaction per wave)
    const float dx = s_rays[rayIdx * 6 + 0];
    const float dy = s_rays[rayIdx * 6 + 1];
    const float dz = s_rays[rayIdx * 6 + 2];
    const float ox = s_rays[rayIdx * 6 + 3];
    const float oy = s_rays[rayIdx * 6 + 4];
    const float oz = s_rays[rayIdx * 6 + 5];

    // ---- ray/AABB slab test (matches reference exactly) --------------------
    const float ivx = 1.0f / ((fabsf(dx) > 1e-9f) ? dx : 1e-9f);
    const float ivy = 1.0f / ((fabsf(dy) > 1e-9f) ? dy : 1e-9f);
    const float ivz = 1.0f / ((fabsf(dz) > 1e-9f) ? dz : 1e-9f);
    const float t1x = (-1.0f - ox) * ivx, t2x = (1.0f - ox) * ivx;
    const float t1y = (-1.0f - oy) * ivy, t2y = (1.0f - oy) * ivy;
    const float t1z = (-1.0f - oz) * ivz, t2z = (1.0f - oz) * ivz;
    float near = fmaxf(fmaxf(fminf(t1x, t2x), fminf(t1y, t2y)), fminf(t1z, t2z));
    float far  = fminf(fminf(fmaxf(t1x, t2x), fmaxf(t1y, t2y)), fmaxf(t1z, t2z));
    near = fmaxf(near, 0.2f);
    far  = fmaxf(far, near + 1e-6f);
    const float dt = (far - near) * (1.0f / 128.0f);   // exact: /128 == *2^-7

    // ---- 4 samples per lane ------------------------------------------------
    const int m0 = lane * 4;
    float xs[4], ys[4], zs[4], ts[4], vs[4];
#pragma unroll
    for (int j = 0; j < 4; ++j) {
        const float t = near + ((float)(m0 + j) + 0.5f) * dt;
        const float x = ox + dx * t;
        const float y = oy + dy * t;
        const float z = oz + dz * t;

        int gx = (int)((x + 1.0f) * 64.0f);
        int gy = (int)((y + 1.0f) * 64.0f);
        int gz = (int)((z + 1.0f) * 64.0f);
        gx = gx < 0 ? 0 : (gx > GRID_R - 1 ? GRID_R - 1 : gx);
        gy = gy < 0 ? 0 : (gy > GRID_R - 1 ? GRID_R - 1 : gy);
        gz = gz < 0 ? 0 : (gz > GRID_R - 1 ? GRID_R - 1 : gz);
        const int idx = ((gx << 7) | gy) << 7 | gz;    // (gx*128+gy)*128+gz

        const bool occ    = dgrid[idx] > 0.5f;         // regular-temporal: keep in L2
        const bool inside = (fabsf(x) <= 1.0f) && (fabsf(y) <= 1.0f) && (fabsf(z) <= 1.0f);
        const bool valid  = occ && inside;

        xs[j] = valid ? x : 0.0f;
        ys[j] = valid ? y : 0.0f;
        zs[j] = valid ? z : 0.0f;
        ts[j] = valid ? t : 0.0f;
        vs[j] = valid ? 1.0f : 0.0f;
    }

    // ---- coalesced non-temporal b128 stores (pure streaming output) --------
    const size_t S = (size_t)ray * M_SAMP + m0;        // first sample index

    // xyz: 12 floats = 48B per lane, contiguous across the wave; 16B aligned.
    {
        float* p = out_xyz + S * 3;
        v4f a = { xs[0], ys[0], zs[0], xs[1] };
        v4f b = { ys[1], zs[1], xs[2], ys[2] };
        v4f c = { zs[2], xs[3], ys[3], zs[3] };
        __builtin_nontemporal_store(a, (v4f*)(p + 0));
        __builtin_nontemporal_store(b, (v4f*)(p + 4));
        __builtin_nontemporal_store(c, (v4f*)(p + 8));
    }
    {
        v4f tq = { ts[0], ts[1], ts[2], ts[3] };
        __builtin_nontemporal_store(tq, (v4f*)(out_z + S));
    }
    {
        v4f vq = { vs[0], vs[1], vs[2], vs[3] };
        __builtin_nontemporal_store(vq, (v4f*)(out_valid + S));
    }
}

extern "C" void kernel_launch(void* const* d_in, const int* in_sizes, int n_in,
                              void* d_out, int out_size, void* d_ws, size_t ws_size,
                              hipStream_t stream) {
    (void)in_sizes; (void)n_in; (void)out_size; (void)d_ws; (void)ws_size;
    const float* rays  = (const float*)d_in[0];   // [N, 6]
    const float* dgrid = (const float*)d_in[1];   // [1, 128^3]
    float* out_xyz   = (float*)d_out;                                   // N*M*3
    float* out_z     = out_xyz + (size_t)N_RAYS * M_SAMP * 3;           // N*M
    float* out_valid = out_z   + (size_t)N_RAYS * M_SAMP;               // N*M

    dim3 grid(N_RAYS / RAYS_PER_BLOCK);   // 16384 blocks
    dim3 block(THREADS);
    raymarch_kernel<<<grid, block, 0, stream>>>(rays, dgrid, out_xyz, out_z, out_valid);
}